// Attention_5823975653530
// MI455X (gfx1250) — compile-verified
//
#include <hip/hip_runtime.h>

typedef __attribute__((ext_vector_type(2))) float v2f;
typedef __attribute__((ext_vector_type(4))) float v4f;
typedef __attribute__((ext_vector_type(8))) float v8f;

#define BB 4
#define HH 16
#define TT1 1024
#define TT2 1024
#define DKK 64
#define DVV 64
#define SPITCH 1032   // 1024 + 8 floats pad (row start stays 16B aligned)

__device__ __forceinline__ float fast_tanh(float x) {
#if __has_builtin(__builtin_amdgcn_tanhf)
    return __builtin_amdgcn_tanhf(x);      // native v_tanh_f32 (gfx1250)
#else
    return tanhf(x);
#endif
}

// exp for bounded arguments (x <= 0, |x| small): single v_exp_f32
__device__ __forceinline__ float fast_exp(float x) {
#if __has_builtin(__builtin_amdgcn_exp2f)
    return __builtin_amdgcn_exp2f(x * 1.4426950408889634f);
#else
    return __expf(x);
#endif
}

__global__ __launch_bounds__(256, 1) void attn_tanh_softmax_kernel(
    const float* __restrict__ Qg, const float* __restrict__ Kg,
    const float* __restrict__ Vg, const float* __restrict__ Mg,
    float* __restrict__ Og, float* __restrict__ Wg)
{
    __shared__ float scores[16][SPITCH];   // 66,048 B score slab
    __shared__ float wpart[8][16];         // per-wave row partials
    __shared__ float rowred[16];           // row max
    __shared__ float rowinv[16];           // 1/rowsum
    __shared__ float pvpart[4][256];       // PV split-K combine

    const int tid  = threadIdx.x;
    const int lane = tid & 31;
    const int wave = tid >> 5;
    const int half = lane >> 4;   // 0: lanes 0-15, 1: lanes 16-31
    const int l16  = lane & 15;

    const int qt = blockIdx.x;    // 0..63 query tiles
    const int h  = blockIdx.y;
    const int b  = blockIdx.z;
    const int q0 = qt * 16;

    const float* Qbase = Qg + (((size_t)(b * HH + h)) * TT1 + q0) * DKK;
    const float* Kbase = Kg + (size_t)b * TT2 * DKK;
    const float* Vbase = Vg + (size_t)b * TT2 * DVV;
    float*       Wbase = Wg + (((size_t)(b * HH + h)) * TT1 + q0) * TT2;
    float*       Obase = Og + (((size_t)(b * HH + h)) * TT1 + q0) * DVV;

    // ---- Preload Q tile A-fragments (16 steps of 16x4) into registers ----
    // step s covers dims [4s,4s+4): lane holds Q[row=l16][4s + 2*half + {0,1}]
    v2f afrag[16];
    {
        const float* qrow = Qbase + l16 * DKK + 2 * half;
        #pragma unroll
        for (int s = 0; s < 16; ++s)
            afrag[s] = *(const v2f*)(qrow + 4 * s);
    }

    // ---- Phase 1: logits = tanh(Q K^T) + mask for this wave's 128 cols ----
    float rmax[8];
    #pragma unroll
    for (int r = 0; r < 8; ++r) rmax[r] = -1e30f;

    for (int t = 0; t < 8; ++t) {
        const int kk0 = wave * 128 + t * 16;
        const float* krow = Kbase + (size_t)(kk0 + l16) * DKK + 2 * half;
        if (t < 7)
            __builtin_prefetch(krow + 16 * DKK, 0, 1);  // next key tile
        v8f acc = {};
        #pragma unroll
        for (int s = 0; s < 16; ++s) {
            v2f bfrag = *(const v2f*)(krow + 4 * s);
            acc = __builtin_amdgcn_wmma_f32_16x16x4_f32(
                false, afrag[s], false, bfrag, (short)0, acc, false, false);
        }
        #pragma unroll
        for (int r = 0; r < 8; ++r) {
            const int row = r + 8 * half;           // M
            const int col = kk0 + l16;              // N
            float v = fast_tanh(acc[r]) + Mg[(size_t)(q0 + row) * TT2 + col];
            scores[row][col] = v;
            rmax[r] = fmaxf(rmax[r], v);
        }
    }

    // ---- Row max: reduce within 16-lane halves, then across waves ----
    #pragma unroll
    for (int r = 0; r < 8; ++r) {
        float m = rmax[r];
        #pragma unroll
        for (int off = 1; off < 16; off <<= 1)
            m = fmaxf(m, __shfl_xor(m, off, 16));
        if (l16 == 0) wpart[wave][r + 8 * half] = m;
    }
    __syncthreads();
    if (tid < 16) {
        float m = wpart[0][tid];
        for (int w = 1; w < 8; ++w) m = fmaxf(m, wpart[w][tid]);
        rowred[tid] = m;
    }
    __syncthreads();

    // ---- exp(x - rowmax), accumulate per-wave row sums ----
    for (int m = 0; m < 16; ++m) {
        const float rm = rowred[m];
        float s_local = 0.f;
        #pragma unroll
        for (int i = 0; i < 4; ++i) {
            const int col = wave * 128 + i * 32 + lane;
            float e = fast_exp(scores[m][col] - rm);
            scores[m][col] = e;
            s_local += e;
        }
        #pragma unroll
        for (int off = 16; off >= 1; off >>= 1)
            s_local += __shfl_xor(s_local, off, 32);
        if (lane == 0) wpart[wave][m] = s_local;
    }
    __syncthreads();
    if (tid < 16) {
        float s = 0.f;
        for (int w = 0; w < 8; ++w) s += wpart[w][tid];
        rowinv[tid] = 1.0f / s;
    }
    __syncthreads();

    // ---- Normalize; stream weights to global (NT b128) + keep in LDS ----
    for (int j = 0; j < 16; ++j) {
        const int i4  = tid + j * 256;      // 0..4095 float4 slots
        const int row = i4 >> 8;            // 256 float4 per row
        const int c4  = (i4 & 255) * 4;
        const float inv = rowinv[row];
        v4f v4 = *(v4f*)&scores[row][c4];
        v4 *= inv;
        *(v4f*)&scores[row][c4] = v4;
        // weights are written once, never re-read from global: NT store keeps L2 for K/V
        __builtin_nontemporal_store(v4, (v4f*)&Wbase[(size_t)row * TT2 + c4]);
    }
    __syncthreads();

    // ---- Phase 2: out = weights @ V ; waves split K=1024 in halves ----
    const int ntile = wave & 3;
    const int n0 = ntile * 16;
    const int k0 = (wave >> 2) * 512;
    v8f acc = {};
    const float* vcol = Vbase + n0 + l16;
    for (int s = 0; s < 128; ++s) {
        const int kk = k0 + 4 * s + 2 * half;
        v2f a = *(const v2f*)&scores[l16][kk];
        v2f bf;
        bf.x = vcol[(size_t)kk * DVV];
        bf.y = vcol[(size_t)(kk + 1) * DVV];
        acc = __builtin_amdgcn_wmma_f32_16x16x4_f32(
            false, a, false, bf, (short)0, acc, false, false);
    }
    if (wave >= 4) {
        #pragma unroll
        for (int r = 0; r < 8; ++r) pvpart[ntile][r * 32 + lane] = acc[r];
    }
    __syncthreads();
    if (wave < 4) {
        #pragma unroll
        for (int r = 0; r < 8; ++r) {
            const float v = acc[r] + pvpart[ntile][r * 32 + lane];
            const int row = r + 8 * half;
            __builtin_nontemporal_store(v, &Obase[(size_t)row * DVV + n0 + l16]);
        }
    }
}

extern "C" void kernel_launch(void* const* d_in, const int* in_sizes, int n_in,
                              void* d_out, int out_size, void* d_ws, size_t ws_size,
                              hipStream_t stream) {
    const float* Q  = (const float*)d_in[0];
    const float* K  = (const float*)d_in[1];
    const float* V  = (const float*)d_in[2];
    const float* Mk = (const float*)d_in[3];
    float* Out = (float*)d_out;                                  // [B,H,T1,DV]
    float* W   = Out + (size_t)BB * HH * TT1 * DVV;              // [B,H,T1,T2]
    dim3 grid(TT1 / 16, HH, BB);
    attn_tanh_softmax_kernel<<<grid, 256, 0, stream>>>(Q, K, V, Mk, Out, W);
}